// Primitive_49555332661845
// MI455X (gfx1250) — compile-verified
//
#include <hip/hip_runtime.h>
#include <hip/hip_bf16.h>

// ---------------------------------------------------------------------------
// Apply a 4x4 gate to qubits (5,12) of a 24-qubit float32 state, batch=4.
//
// Memory layout (element index, 26 bits, MSB->LSB):
//   [q0..q4 (5b) | q5 | q6..q11 (6b) | q12 | q13..q23 (11b) | batch (2b)]
// q5 bit -> +2^20 elements, q12 bit -> +2^13 elements.
//
// For each of 2^24 columns n (non-target bits + batch):  Y[o] = sum_k M[o][k] X[k]
//   k = 2*b5_in + b12_in, o = 2*b5_out + b12_out, M[o][k] = pauli[4*o + k].
//
// Bandwidth bound (512 MB traffic, ~1 FLOP/byte -> ~22 us at 23.3 TB/s).
// WMMA mapping per wave per 64-column group:
//   A (16x4 f32): rows 0..3 = gate M, rows 4..15 = 0
//   B (4x16 f32): 16 state columns; float4 component j feeds WMMA j
//   D (16x16)   : rows 0..3 = results
// ---------------------------------------------------------------------------

typedef float v2f __attribute__((ext_vector_type(2)));
typedef float v8f __attribute__((ext_vector_type(8)));

#define HAVE_WMMA_F32_16X16X4 __has_builtin(__builtin_amdgcn_wmma_f32_16x16x4_f32)

// Address-bit constants for this problem size / target pair.
static constexpr uint32_t K12_OFF = 1u << 13;  // q12 bit (element stride)
static constexpr uint32_t K5_OFF  = 1u << 20;  // q5  bit (element stride)
static constexpr uint32_t LOW_MASK = (1u << 13) - 1;  // bits below q12
static constexpr uint32_t MID_MASK = 63u;             // q6..q11 (6 bits)

__device__ __forceinline__ uint32_t group_base00(uint32_t g) {
  // Column index base for group g (64 columns/group) -> element offset with
  // k5 = k12 = 0.
  uint32_t nb  = g << 6;
  uint32_t low = nb & LOW_MASK;
  uint32_t mid = (nb >> 13) & MID_MASK;
  uint32_t hi  = nb >> 19;
  return (hi << 21) | (mid << 14) | low;
}

__global__ __launch_bounds__(256) void pauli2q_apply(
    const float* __restrict__ state,
    const float* __restrict__ pauli,
    float* __restrict__ out,
    int n_groups)
{
  const int lane = threadIdx.x & 31;
  const int wavesPerBlock = blockDim.x >> 5;
  const int wave   = blockIdx.x * wavesPerBlock + (threadIdx.x >> 5);
  const int nwaves = gridDim.x * wavesPerBlock;

#if HAVE_WMMA_F32_16X16X4
  // ---- A matrix (16x4 f32, 2 VGPRs): lanes 0-15 hold K={0,1}, lanes 16-31
  //      hold K={2,3} for the same rows M = lane&15. Rows >= 4 are zero.
  const int m = lane & 15;
  v2f a = {};
  if (m < 4) {
    const float* p = pauli + m * 4 + ((lane < 16) ? 0 : 2);
    a.x = p[0];
    a.y = p[1];
  }

  const uint32_t c0    = 4u * (uint32_t)m;                 // 4 columns per lane
  const uint32_t k5off = (lane < 16) ? 0u : K5_OFF;        // lane-half -> k5 bit

  for (int g = wave; g < n_groups; g += nwaves) {
    const uint32_t base00 = group_base00((uint32_t)g);

    // Row pair (k, k+1) for this lane-half; 4 consecutive columns each.
    const float4 in0 = *(const float4*)(state + base00 + k5off + c0);            // k=0 / k=2
    const float4 in1 = *(const float4*)(state + base00 + k5off + K12_OFF + c0);  // k=1 / k=3

    const v8f cz = {};
    v2f b;

    b.x = in0.x; b.y = in1.x;   // columns 4*nl + 0
    v8f d0 = __builtin_amdgcn_wmma_f32_16x16x4_f32(false, a, false, b, (short)0, cz, false, false);
    b.x = in0.y; b.y = in1.y;   // columns 4*nl + 1
    v8f d1 = __builtin_amdgcn_wmma_f32_16x16x4_f32(false, a, false, b, (short)0, cz, false, false);
    b.x = in0.z; b.y = in1.z;   // columns 4*nl + 2
    v8f d2 = __builtin_amdgcn_wmma_f32_16x16x4_f32(false, a, false, b, (short)0, cz, false, false);
    b.x = in0.w; b.y = in1.w;   // columns 4*nl + 3
    v8f d3 = __builtin_amdgcn_wmma_f32_16x16x4_f32(false, a, false, b, (short)0, cz, false, false);

    // D rows 0..3 live in VGPRs 0..3, lanes 0..15 (N = lane). Lane l stores
    // the float4 of columns 4l..4l+3 for each output row r.
    if (lane < 16) {
      float4 o;
      o.x = d0[0]; o.y = d1[0]; o.z = d2[0]; o.w = d3[0];
      *(float4*)(out + base00 + c0) = o;                       // o=0: b5=0,b12=0
      o.x = d0[1]; o.y = d1[1]; o.z = d2[1]; o.w = d3[1];
      *(float4*)(out + base00 + K12_OFF + c0) = o;             // o=1: b12=1
      o.x = d0[2]; o.y = d1[2]; o.z = d2[2]; o.w = d3[2];
      *(float4*)(out + base00 + K5_OFF + c0) = o;              // o=2: b5=1
      o.x = d0[3]; o.y = d1[3]; o.z = d2[3]; o.w = d3[3];
      *(float4*)(out + base00 + K5_OFF + K12_OFF + c0) = o;    // o=3: both
    }
  }
#else
  // Scalar fallback (still coalesced): each lane handles 2 columns per group.
  float M[16];
  #pragma unroll
  for (int i = 0; i < 16; ++i) M[i] = pauli[i];

  for (int g = wave; g < n_groups; g += nwaves) {
    const uint32_t base00 = group_base00((uint32_t)g);
    const uint32_t cc = 2u * (uint32_t)lane;
    float2 x0 = *(const float2*)(state + base00 + cc);
    float2 x1 = *(const float2*)(state + base00 + K12_OFF + cc);
    float2 x2 = *(const float2*)(state + base00 + K5_OFF + cc);
    float2 x3 = *(const float2*)(state + base00 + K5_OFF + K12_OFF + cc);
    #pragma unroll
    for (int o = 0; o < 4; ++o) {
      float2 y;
      y.x = M[4*o+0]*x0.x + M[4*o+1]*x1.x + M[4*o+2]*x2.x + M[4*o+3]*x3.x;
      y.y = M[4*o+0]*x0.y + M[4*o+1]*x1.y + M[4*o+2]*x2.y + M[4*o+3]*x3.y;
      uint32_t ob = base00 + ((uint32_t)(o >> 1) << 20) + ((uint32_t)(o & 1) << 13) + cc;
      *(float2*)(out + ob) = y;
    }
  }
#endif
}

extern "C" void kernel_launch(void* const* d_in, const int* in_sizes, int n_in,
                              void* d_out, int out_size, void* d_ws, size_t ws_size,
                              hipStream_t stream) {
  const float* state = (const float*)d_in[0];   // 2^26 floats
  const float* pauli = (const float*)d_in[1];   // 16 floats (4x4 gate)
  float* out = (float*)d_out;                   // 2^26 floats

  // columns = elements/4 (k-dim), 64 columns per group.
  int n_groups = in_sizes[0] >> 8;              // 2^18 for the reference size
  if (n_groups < 1) n_groups = 1;

  const int threads = 256;                       // 8 waves per block
  const int wavesPerBlock = threads / 32;
  const int groupsPerWave = 4;                   // grid-stride depth
  int blocks = (n_groups + wavesPerBlock * groupsPerWave - 1) /
               (wavesPerBlock * groupsPerWave);
  if (blocks < 1) blocks = 1;

  pauli2q_apply<<<blocks, threads, 0, stream>>>(state, pauli, out, n_groups);
}